// LoRALinear_18683107738116
// MI455X (gfx1250) — compile-verified
//
#include <hip/hip_runtime.h>

#define LORA_SCALE 2.0f   // alpha / r = 32/16
#define R_DIM 16

typedef float v2f __attribute__((ext_vector_type(2)));
typedef float v8f __attribute__((ext_vector_type(8)));
typedef int   v4i __attribute__((ext_vector_type(4)));

#if defined(__has_builtin)
#  if __has_builtin(__builtin_amdgcn_global_load_async_to_lds_b128)
#    define HAVE_ASYNC_LDS 1
#  endif
#endif
#ifndef HAVE_ASYNC_LDS
#  define HAVE_ASYNC_LDS 0
#endif

#define AS_GLOBAL __attribute__((address_space(1)))
#define AS_LDS    __attribute__((address_space(3)))

// ---------------------------------------------------------------------------
// Kernel 1: T[m][r] = LORA_SCALE * sum_k x[m][k] * A[k][r]     (A is [K][16])
// ---------------------------------------------------------------------------
__global__ __launch_bounds__(256) void lora_xa_kernel(
    const float* __restrict__ x, const float* __restrict__ A,
    float* __restrict__ T, int M, int K) {
  const int lane = threadIdx.x & 31;
  const int row  = (int)((blockIdx.x * blockDim.x + threadIdx.x) >> 5);
  if (row >= M) return;

  const float* xr = x + (size_t)row * K;

  float acc[R_DIM];
#pragma unroll
  for (int r = 0; r < R_DIM; ++r) acc[r] = 0.0f;

  for (int k0 = lane * 4; k0 < K; k0 += 128) {
    float4 xv = *(const float4*)(xr + k0);
    const float* ab = A + (size_t)k0 * R_DIM;
#pragma unroll
    for (int j = 0; j < 4; ++j) {
      float xs = (&xv.x)[j];
#pragma unroll
      for (int q = 0; q < 4; ++q) {
        float4 av = *(const float4*)(ab + j * R_DIM + q * 4);
        acc[q * 4 + 0] = fmaf(xs, av.x, acc[q * 4 + 0]);
        acc[q * 4 + 1] = fmaf(xs, av.y, acc[q * 4 + 1]);
        acc[q * 4 + 2] = fmaf(xs, av.z, acc[q * 4 + 2]);
        acc[q * 4 + 3] = fmaf(xs, av.w, acc[q * 4 + 3]);
      }
    }
  }

#pragma unroll
  for (int r = 0; r < R_DIM; ++r) {
    float v = acc[r];
#pragma unroll
    for (int off = 16; off > 0; off >>= 1) v += __shfl_xor(v, off, 32);
    acc[r] = v * LORA_SCALE;
  }

  if (lane == 0) {
#pragma unroll
    for (int r = 0; r < R_DIM; r += 4) {
      float4 o = make_float4(acc[r], acc[r + 1], acc[r + 2], acc[r + 3]);
      *(float4*)(T + (size_t)row * R_DIM + r) = o;
    }
  }
}

// ---------------------------------------------------------------------------
// Kernel 2: out = x @ W^T + b + T @ lora_B
// 128x128 block tile, BK=32, 256 threads = 8 waves (2Mx4N), 8 f32 WMMA accs
// per wave. Double-buffered LDS fed by GLOBAL_LOAD_ASYNC_TO_LDS_B128 with an
// XOR chunk swizzle; fragment registers double-buffered across K-substeps.
// ---------------------------------------------------------------------------
#define BM 128
#define BN 128
#define BK 32
#define TSTR 20   // epilogue LDS row stride for 16-wide tiles

__device__ __forceinline__ void wait_async_le8() {
#if HAVE_ASYNC_LDS
#if defined(__has_builtin) && __has_builtin(__builtin_amdgcn_s_wait_asynccnt)
  __builtin_amdgcn_s_wait_asynccnt(8);
#else
  asm volatile("s_wait_asynccnt 0x8" ::: "memory");
#endif
#endif
}

__device__ __forceinline__ void wait_async_le0() {
#if HAVE_ASYNC_LDS
#if defined(__has_builtin) && __has_builtin(__builtin_amdgcn_s_wait_asynccnt)
  __builtin_amdgcn_s_wait_asynccnt(0);
#else
  asm volatile("s_wait_asynccnt 0x0" ::: "memory");
#endif
#endif
}

// Stage one BK-slice of X and W into (swizzled) LDS buffers.
__device__ __forceinline__ void issue_tile_loads(
    const float* __restrict__ gx, const float* __restrict__ gw,
    float* ldsA, float* ldsW, const int* rows, const int* c4s,
    int K, int k0) {
#pragma unroll
  for (int v = 0; v < 4; ++v) {
    const int row = rows[v];
    const int c4  = c4s[v];
    const int sw  = ((c4 ^ ((row >> 1) & 7)) << 2);   // swizzled float offset
#if HAVE_ASYNC_LDS
    __builtin_amdgcn_global_load_async_to_lds_b128(
        (AS_GLOBAL v4i*)(gx + (size_t)row * K + k0 + c4 * 4),
        (AS_LDS v4i*)(ldsA + row * BK + sw), 0, 0);
    __builtin_amdgcn_global_load_async_to_lds_b128(
        (AS_GLOBAL v4i*)(gw + (size_t)row * K + k0 + c4 * 4),
        (AS_LDS v4i*)(ldsW + row * BK + sw), 0, 0);
#else
    float4 xv = *(const float4*)(gx + (size_t)row * K + k0 + c4 * 4);
    float4 wv = *(const float4*)(gw + (size_t)row * K + k0 + c4 * 4);
    *(float4*)(ldsA + row * BK + sw) = xv;
    *(float4*)(ldsW + row * BK + sw) = wv;
#endif
  }
}

// Load one K-substep's A/B fragments from (swizzled) LDS.
__device__ __forceinline__ void load_frags(
    const float* curA, const float* curW,
    int rowBaseA, int rowBaseW, int kk, int s0, v2f* a, v2f* b) {
  const int c = ((kk ^ s0) << 2);
#pragma unroll
  for (int mt = 0; mt < 4; ++mt)
    a[mt] = *(const v2f*)(curA + rowBaseA + mt * 16 * BK + c);
#pragma unroll
  for (int nt = 0; nt < 2; ++nt)
    b[nt] = *(const v2f*)(curW + rowBaseW + nt * 16 * BK + c);
}

__global__ __launch_bounds__(256) void lora_gemm_kernel(
    const float* __restrict__ x, const float* __restrict__ W,
    const float* __restrict__ bias, const float* __restrict__ loraB,
    const float* __restrict__ T, float* __restrict__ out,
    int M, int N, int K) {
  __shared__ float lds_x[2][BM * BK];   // 2 x 16KB
  __shared__ float lds_w[2][BN * BK];   // 2 x 16KB

  const int tid  = threadIdx.x;
  const int lane = tid & 31;
  const int wid  = tid >> 5;     // 0..7
  const int wm   = wid >> 2;     // 0..1  -> 64 rows
  const int wn   = wid & 3;      // 0..3  -> 32 cols
  const int half = lane >> 4;    // K-half of fragments
  const int l16  = lane & 15;
  const int s0   = (l16 >> 1) & 7;   // per-lane swizzle constant

  const int rowBaseA = (wm * 64 + l16) * BK + 2 * half;
  const int rowBaseW = (wn * 32 + l16) * BK + 2 * half;

  const int bm = blockIdx.y * BM;
  const int bn = blockIdx.x * BN;

  const float* gx = x + (size_t)bm * K;
  const float* gw = W + (size_t)bn * K;

  v8f acc[4][2] = {};

  // each thread stages 4 b128 chunks per tile (1024 chunks / 256 threads)
  int rows[4], c4s[4];
#pragma unroll
  for (int v = 0; v < 4; ++v) {
    int li = tid + 256 * v;
    rows[v] = li >> 3;   // 0..127
    c4s[v]  = li & 7;    // 0..7
  }

  issue_tile_loads(gx, gw, lds_x[0], lds_w[0], rows, c4s, K, 0);

  const int nStages = K / BK;
  for (int s = 0; s < nStages; ++s) {
    const float* curA = lds_x[s & 1];
    const float* curW = lds_w[s & 1];

    if (s + 1 < nStages) {
      issue_tile_loads(gx, gw, lds_x[(s + 1) & 1], lds_w[(s + 1) & 1],
                       rows, c4s, K, (s + 1) * BK);
      wait_async_le8();    // oldest batch (stage s) has landed in LDS
    } else {
      wait_async_le0();
    }
    __syncthreads();

    // 8 K-substeps of 4; fragment regs double-buffered so sub-step k+1's DS
    // loads are in flight during sub-step k's 8 WMMAs.
    v2f a[2][4], b[2][2];
    load_frags(curA, curW, rowBaseA, rowBaseW, 0, s0, a[0], b[0]);
#pragma unroll
    for (int kk = 0; kk < 8; ++kk) {
      const int cur = kk & 1;
      const int nxt = cur ^ 1;
      if (kk < 7)
        load_frags(curA, curW, rowBaseA, rowBaseW, kk + 1, s0, a[nxt], b[nxt]);
#pragma unroll
      for (int mt = 0; mt < 4; ++mt)
#pragma unroll
        for (int nt = 0; nt < 2; ++nt)
          acc[mt][nt] = __builtin_amdgcn_wmma_f32_16x16x4_f32(
              false, a[cur][mt], false, b[cur][nt], (short)0, acc[mt][nt],
              false, false);
    }
    __syncthreads();   // protect buffer overwritten by next iteration's issue
  }

  // ------------------------- LoRA epilogue (WMMA, K=16) -------------------
  float* eT = &lds_x[0][0];   // T block [128][16], stride TSTR
  float* eB = &lds_w[0][0];   // lora_B block transposed [n][r], stride TSTR
  {
    const int rowT = tid >> 1;
    const int off  = (tid & 1) * 8;
    float4 t0 = *(const float4*)(T + (size_t)(bm + rowT) * R_DIM + off);
    float4 t1 = *(const float4*)(T + (size_t)(bm + rowT) * R_DIM + off + 4);
    *(float4*)(eT + rowT * TSTR + off)     = t0;
    *(float4*)(eT + rowT * TSTR + off + 4) = t1;
  }
  {
    const int n  = tid >> 1;
    const int rb = (tid & 1) * 8;
#pragma unroll
    for (int j = 0; j < 8; ++j)
      eB[n * TSTR + rb + j] = loraB[(size_t)(rb + j) * N + bn + n];
  }
  __syncthreads();

#pragma unroll
  for (int kk = 0; kk < 4; ++kk) {
    v2f a[4], b[2];
#pragma unroll
    for (int mt = 0; mt < 4; ++mt)
      a[mt] = *(const v2f*)(eT + (wm * 64 + mt * 16 + l16) * TSTR + kk * 4 + 2 * half);
#pragma unroll
    for (int nt = 0; nt < 2; ++nt)
      b[nt] = *(const v2f*)(eB + (wn * 32 + nt * 16 + l16) * TSTR + kk * 4 + 2 * half);
#pragma unroll
    for (int mt = 0; mt < 4; ++mt)
#pragma unroll
      for (int nt = 0; nt < 2; ++nt)
        acc[mt][nt] = __builtin_amdgcn_wmma_f32_16x16x4_f32(
            false, a[mt], false, b[nt], (short)0, acc[mt][nt], false, false);
  }

  // ------------------------- bias + store --------------------------------
  float bv[2];
#pragma unroll
  for (int nt = 0; nt < 2; ++nt) bv[nt] = bias[bn + wn * 32 + nt * 16 + l16];

#pragma unroll
  for (int mt = 0; mt < 4; ++mt) {
#pragma unroll
    for (int nt = 0; nt < 2; ++nt) {
      const int col   = bn + wn * 32 + nt * 16 + l16;
      const int rbase = bm + wm * 64 + mt * 16 + 8 * half;
#pragma unroll
      for (int i = 0; i < 8; ++i)
        out[(size_t)(rbase + i) * N + col] = acc[mt][nt][i] + bv[nt];
    }
  }
}

// ---------------------------------------------------------------------------
extern "C" void kernel_launch(void* const* d_in, const int* in_sizes, int n_in,
                              void* d_out, int out_size, void* d_ws, size_t ws_size,
                              hipStream_t stream) {
  (void)n_in; (void)out_size; (void)ws_size;
  const float* x      = (const float*)d_in[0];
  const float* W      = (const float*)d_in[1];
  const float* b      = (const float*)d_in[2];
  const float* lora_A = (const float*)d_in[3];
  const float* lora_B = (const float*)d_in[4];
  float* out = (float*)d_out;

  const int D_out = in_sizes[2];                 // 4096
  const int R     = in_sizes[4] / D_out;         // 16
  const int D_in  = in_sizes[3] / R;             // 4096
  const int M     = (int)((long long)in_sizes[0] / D_in);  // B*S = 16384

  float* T = (float*)d_ws;                       // [M][R]

  dim3 g1((unsigned)((M + 7) / 8));
  lora_xa_kernel<<<g1, 256, 0, stream>>>(x, lora_A, T, M, D_in);

  dim3 g2((unsigned)(D_out / BN), (unsigned)(M / BM));
  lora_gemm_kernel<<<g2, 256, 0, stream>>>(x, W, b, lora_B, T, out, M, D_out, D_in);
}